// Task_Decompose_10934986735975
// MI455X (gfx1250) — compile-verified
//
#include <hip/hip_runtime.h>
#include <stdint.h>

// ---------------------------------------------------------------------------
// Task: pure gather/concat (no matmul). Bandwidth-bound: ~209MB NT writes,
// gather sources (gf 1MB, cf 2MB, tables) are L2-resident on MI455X (192MB L2).
// Floor ~9us at 23.3 TB/s. CDNA5 paths used: async global->LDS staging
// (ASYNCcnt), global_prefetch_b8, nontemporal B128 stores.
// ---------------------------------------------------------------------------

typedef float f32x4 __attribute__((ext_vector_type(4)));
typedef int   v4i   __attribute__((ext_vector_type(4)));

#define B_   4
#define P_   1024
#define M_   12
#define N_   256
#define L_   512
#define D_   256      // graph feature dim
#define DC_  256      // context feature dim
#define ROWF 1064     // floats per output feature row: 512 + 20 + 20 + 512
#define FEA_TOTAL ((size_t)B_ * P_ * M_ * ROWF)

// ----- CDNA5 async global->LDS (guarded; fallback = sync copy via LDS) -----
#if defined(__has_builtin)
#  if __has_builtin(__builtin_amdgcn_global_load_async_to_lds_b128)
#    define HAVE_ASYNC_B128 1
#  endif
#  if __has_builtin(__builtin_amdgcn_s_wait_asynccnt)
#    define HAVE_WAIT_ASYNC_BUILTIN 1
#  endif
#endif

#ifdef HAVE_ASYNC_B128
// Param 0 is a v4i pointer in AS(1) ("__device__"); assume AS(3) v4i* for LDS.
#  define ASYNC16(gsrc, ldst)                                                  \
     __builtin_amdgcn_global_load_async_to_lds_b128(                           \
         (__attribute__((address_space(1))) v4i*)(void*)(gsrc),                \
         (__attribute__((address_space(3))) v4i*)(void*)(ldst), 0, 0)
#else
#  define ASYNC16(gsrc, ldst) (*(ldst) = *(gsrc))
#endif

__device__ __forceinline__ void wait_async_all() {
#ifdef HAVE_ASYNC_B128
#  ifdef HAVE_WAIT_ASYNC_BUILTIN
  __builtin_amdgcn_s_wait_asynccnt(0);
#  else
  asm volatile("s_wait_asynccnt 0" ::: "memory");
#  endif
#endif
}

__global__ __launch_bounds__(256) void decompose_kernel(
    const int*   __restrict__ rp,   // relation_path  [B,P,12,4] int32
    const int*   __restrict__ pi,   // path_info      [B,N,6]    int32
    const float* __restrict__ gf,   // graph_feature  [B,N,256]
    const float* __restrict__ cf,   // context_feature[B,L,256]
    const float* __restrict__ dw,   // dis_embed_w      [20,20]
    const float* __restrict__ sw,   // dis_sent_embed_w [20,20]
    float*       __restrict__ out)  // [B,P,12,1064] fea ++ [B,P,12] mask
{
  __shared__ f32x4 s_gf[M_ * 128];           // 24KB: per row m, [ia row | ib row]
  __shared__ int   s_ia[M_], s_ib[M_], s_idx[M_];
  __shared__ int   s_c0[M_], s_c1[M_], s_c2[M_], s_c3[M_];
  __shared__ float s_mask[M_];

  const int bp  = blockIdx.x;                // b*P + p  (P == 1024)
  const int b   = bp >> 10;
  const int tid = threadIdx.x;

  // ---- phase 1: per-row scalars (indices, distance bucket, mask) ----
  if (tid < M_) {
    const int m  = tid;
    const int* q = rp + (((size_t)bp) * M_ + m) * 4;
    const int s0 = q[0], s1 = q[1], s2 = q[2], s3 = q[3];
    const int ib = (m < 4) ? s2 : s3;        // gb = 2 for branch1, 3 otherwise
    const int pibase = b * N_ * 6;
    const int p0 = pi[pibase + s0 * 6];      // pos[sel0] (== pos of ia)
    const int pb = pi[pibase + ib * 6];
    const int p2 = pi[pibase + s2 * 6];
    const int delta = p0 - pb;
    const int v  = delta & 511;              // jnp.mod(delta, 512)
    const int d2 = v ? (32 - __builtin_clz((unsigned)v)) : 0;  // DIS2IDX
    s_idx[m]  = (delta < 0) ? (10 - d2) : (10 + d2);
    s_ia[m]   = s0;
    s_ib[m]   = ib;
    s_c0[m]   = p0;
    s_c2[m]   = p2;
    s_c1[m]   = (m >= 8) ? pi[pibase + s1 * 6] : 0;
    s_c3[m]   = (m >= 8) ? pi[pibase + s3 * 6] : 0;
    s_mask[m] = ((s0 | s1 | s2 | s3) != 0) ? 1.0f : 0.0f;  // sel >= 0 always
  }
  __syncthreads();

  const f32x4* gf4 = (const f32x4*)(gf + (size_t)b * N_ * D_);   // 64 f32x4/row
  const f32x4* cf4 = (const f32x4*)(cf + (size_t)b * L_ * DC_);  // 64 f32x4/row
  float* ofea = out + (size_t)bp * M_ * ROWF;

  // ---- phase 2: async-stage all 24 gf gather rows into LDS ----
  for (int t = tid; t < M_ * 128; t += 256) {
    const int m = t >> 7, q = t & 127;
    const int n = (q < 64) ? s_ia[m] : s_ib[m];
    const f32x4* src = gf4 + (size_t)n * 64 + (q & 63);
    ASYNC16(src, &s_gf[t]);
  }
  // prefetch cf gather rows while async copies fly
  if (tid < M_) {
    __builtin_prefetch(cf4 + (size_t)s_c0[tid] * 64, 0, 3);
    __builtin_prefetch(cf4 + (size_t)s_c2[tid] * 64, 0, 3);
    if (tid >= 8) {
      __builtin_prefetch(cf4 + (size_t)s_c1[tid] * 64, 0, 3);
      __builtin_prefetch(cf4 + (size_t)s_c3[tid] * 64, 0, 3);
    }
  }

  // ---- phase 3a: emb + semb (floats 512..551 of each row) ----
  for (int t = tid; t < M_ * 40; t += 256) {
    const int m = t / 40, c = t % 40;
    const int idx = s_idx[m];
    const float val = (c < 20) ? dw[idx * 20 + c] : sw[idx * 20 + (c - 20)];
    __builtin_nontemporal_store(val * s_mask[m],
                                ofea + (size_t)m * ROWF + 512 + c);
  }

  // ---- phase 3b: con section (floats 552..1063), direct L2 gathers ----
  for (int t = tid; t < M_ * 128; t += 256) {
    const int m = t >> 7, q = t & 127, qq = q & 63;
    f32x4 val;
    if (q < 64) {
      val = cf4[(size_t)s_c0[m] * 64 + qq];
      if (m >= 8) val += cf4[(size_t)s_c1[m] * 64 + qq];   // logical branch
    } else {
      val = cf4[(size_t)s_c2[m] * 64 + qq];
      if (m >= 8) val += cf4[(size_t)s_c3[m] * 64 + qq];
    }
    val *= s_mask[m];
    f32x4* orow = (f32x4*)(ofea + (size_t)m * ROWF);        // 16B-aligned
    __builtin_nontemporal_store(val, orow + 138 + q);       // 552/4 == 138
  }

  // ---- phase 4: rep section (floats 0..511) streamed out of LDS ----
  wait_async_all();
  __syncthreads();
  for (int t = tid; t < M_ * 128; t += 256) {
    const int m = t >> 7, q = t & 127;
    const f32x4 val = s_gf[t] * s_mask[m];
    f32x4* orow = (f32x4*)(ofea + (size_t)m * ROWF);
    __builtin_nontemporal_store(val, orow + q);
  }

  // ---- phase 5: mask output (same condition as zero_mask) ----
  if (tid < M_) {
    out[FEA_TOTAL + (size_t)bp * M_ + tid] = s_mask[tid];
  }
}

extern "C" void kernel_launch(void* const* d_in, const int* in_sizes, int n_in,
                              void* d_out, int out_size, void* d_ws, size_t ws_size,
                              hipStream_t stream) {
  (void)in_sizes; (void)n_in; (void)out_size; (void)d_ws; (void)ws_size;
  const int*   rp = (const int*)  d_in[0];  // relation_path
  const int*   pi = (const int*)  d_in[1];  // path_info
  const float* gf = (const float*)d_in[2];  // graph_feature
  const float* cf = (const float*)d_in[3];  // context_feature
  const float* dw = (const float*)d_in[4];  // dis_embed_w
  const float* sw = (const float*)d_in[5];  // dis_sent_embed_w
  float* out = (float*)d_out;

  decompose_kernel<<<dim3(B_ * P_), dim3(256), 0, stream>>>(
      rp, pi, gf, cf, dw, sw, out);
}